// MedicalSSIMLoss_5394478924574
// MI455X (gfx1250) — compile-verified
//
#include <hip/hip_runtime.h>

typedef __attribute__((ext_vector_type(2))) float v2f;
typedef __attribute__((ext_vector_type(8))) float v8f;
typedef __attribute__((ext_vector_type(4))) unsigned int u32x4;
typedef __attribute__((ext_vector_type(8))) int i32x8;
typedef __attribute__((ext_vector_type(4))) int i32x4;

#define WIN    7
#define HV     512
#define WV     512
#define OUTD   506               // 512 - 6 valid outputs per dim
#define TILE_R 32                // output rows per block
#define TILE_C 64                // output cols per block
#define IN_R   (TILE_R + WIN - 1)  // 38
#define IN_C   (TILE_C + WIN - 1)  // 70 (used cols; TDM fetches 72)
#define LDS_W  72                // LDS row stride in dwords == TDM tile_dim0
#define TILES_R ((OUTD + TILE_R - 1) / TILE_R)   // 16
#define TILES_C ((OUTD + TILE_C - 1) / TILE_C)   // 8
#define NBATCH 64
#define NBLOCKS (NBATCH * TILES_R * TILES_C)     // 8192

#if __has_builtin(__builtin_amdgcn_tensor_load_to_lds)
#define USE_TDM 1
#else
#define USE_TDM 0
#endif

#if USE_TDM
// Issue one TDM 2D-tile load: 38 rows x 72 dwords from a row-major f32 tensor
// (row stride 512 dwords) into LDS at lds_byte_off.  D# layout per CDNA5 ISA §8.
__device__ static inline void tdm_load_tile(const float* gtile, unsigned int lds_byte_off) {
    unsigned long long ga = (unsigned long long)(uintptr_t)gtile;

    u32x4 g0;
    g0[0] = 1u;                                    // count=1 (valid), user mode, no gather
    g0[1] = lds_byte_off;                          // lds_addr (bytes)
    g0[2] = (unsigned int)(ga & 0xFFFFFFFFu);      // global_addr[31:0]
    g0[3] = (unsigned int)((ga >> 32) & 0x01FFFFFFu) | (2u << 30); // addr[56:32] | type=2

    i32x8 g1;
    g1[0] = (int)(2u << 16);          // workgroup_mask=0, data_size=2 (4B), no pad/iter
    g1[1] = (int)((unsigned)WV << 16);          // tensor_dim0[15:0]=512 (bits 79:64.. packed)
    g1[2] = (int)((unsigned)(NBATCH * HV) << 16); // tensor_dim1[15:0]=32768 (whole batch)
    g1[3] = (int)((unsigned)LDS_W << 16);       // tensor_dim1 hi=0 | tile_dim0=72
    g1[4] = IN_R;                               // tile_dim1=38, tile_dim2=0
    g1[5] = WV;                                 // tensor_dim0_stride lo = 512
    g1[6] = 0;                                  // stride hi | dim1_stride lo (unused, 2D)
    g1[7] = 0;                                  // dim1_stride hi (unused)

    i32x4 z4 = {0, 0, 0, 0};
#if defined(__clang_major__) && (__clang_major__ >= 23)
    i32x8 z8 = {0, 0, 0, 0, 0, 0, 0, 0};
    __builtin_amdgcn_tensor_load_to_lds(g0, g1, z4, z4, z8, 0);
#else
    __builtin_amdgcn_tensor_load_to_lds(g0, g1, z4, z4, 0);
#endif
}
#endif

__launch_bounds__(256)
__global__ void ssim_tile_kernel(const float* __restrict__ img1,
                                 const float* __restrict__ img2,
                                 float* __restrict__ partial) {
    // Single flat LDS block so x/y tile offsets are known for the TDM descriptor.
    __shared__ __align__(16) float lds_buf[2 * IN_R * LDS_W];
    __shared__ float wsum[8];

    float (*xs)[LDS_W] = (float(*)[LDS_W]) lds_buf;
    float (*ys)[LDS_W] = (float(*)[LDS_W])(lds_buf + IN_R * LDS_W);

    const int bid = blockIdx.x;
    const int tc  = bid % TILES_C;
    const int tr  = (bid / TILES_C) % TILES_R;
    const int b   = bid / (TILES_C * TILES_R);
    const int r0  = tr * TILE_R;
    const int c0  = tc * TILE_C;

    const float* __restrict__ xim = img1 + (size_t)b * HV * WV;
    const float* __restrict__ yim = img2 + (size_t)b * HV * WV;

    const int tid  = threadIdx.x;
    const int wave = tid >> 5;
    const int lane = tid & 31;

#if USE_TDM
    // ---- Tensor Data Mover: DMA both 38x72 halo tiles straight into LDS ----
    if (wave == 0) {
        unsigned int xs_lds = (unsigned int)(uintptr_t)(void*)&xs[0][0]; // LDS off = addr[31:0]
        unsigned int ys_lds = (unsigned int)(uintptr_t)(void*)&ys[0][0];
        tdm_load_tile(xim + (size_t)r0 * WV + c0, xs_lds);
        tdm_load_tile(yim + (size_t)r0 * WV + c0, ys_lds);
        __builtin_amdgcn_s_wait_tensorcnt(0);
    }
    __syncthreads();
#else
    // ---- fallback: cooperative tile + halo load (clamped at image edge) ----
    for (int idx = tid; idx < IN_R * IN_C; idx += 256) {
        int rr = idx / IN_C;
        int cc = idx - rr * IN_C;
        int gr = min(r0 + rr, HV - 1);
        int gc = min(c0 + cc, WV - 1);
        xs[rr][cc] = xim[gr * WV + gc];
        ys[rr][cc] = yim[gr * WV + gc];
    }
    __syncthreads();
#endif

    // ---- wave -> 16x16 output sub-tile (2 rows x 4 cols of waves) ----
    const int wr0  = (wave >> 2) * 16;   // 0 or 16
    const int wc0  = (wave & 3) * 16;    // 0,16,32,48
    const int half = lane >> 4;          // K half-select (A/B layout)
    const int lcol = lane & 15;          // M for A, N for B/C/D

    v8f Sx  = {0,0,0,0,0,0,0,0};
    v8f Sy  = {0,0,0,0,0,0,0,0};
    v8f Sxx = {0,0,0,0,0,0,0,0};
    v8f Syy = {0,0,0,0,0,0,0,0};
    v8f Sxy = {0,0,0,0,0,0,0,0};

    // Out(16x16) = A(16x24 band of ones) x H(24x16 horiz box sums), K chunks of 4.
    #pragma unroll
    for (int q = 0; q < 6; ++q) {
        const int k0 = 4 * q + 2 * half;   // VGPR0 K index for this lane-half
        const int k1 = k0 + 1;             // VGPR1 K index

        // A band chunk: A[m][k] = 1 if m <= k <= m+6  (16x4 f32 layout)
        v2f A;
        A.x = (lcol <= k0 && k0 <= lcol + 6) ? 1.0f : 0.0f;
        A.y = (lcol <= k1 && k1 <= lcol + 6) ? 1.0f : 0.0f;

        // B chunks: rows k0,k1 of the 7-tap horizontal sums at column lcol,
        // for all five maps (x, y, x^2, y^2, xy) from one pass over LDS.
        float bx[2], by[2], bxx[2], byy[2], bxy[2];
        #pragma unroll
        for (int t = 0; t < 2; ++t) {
            const int krow = k0 + t;       // row inside 22-row window (>=22 -> pad)
            float sx = 0.f, sy = 0.f, sxx = 0.f, syy = 0.f, sxy = 0.f;
            if (krow < 22) {
                const int lr = wr0 + krow;
                const int lc = wc0 + lcol;
                #pragma unroll
                for (int k = 0; k < WIN; ++k) {
                    float xv = xs[lr][lc + k];
                    float yv = ys[lr][lc + k];
                    sx += xv;
                    sy += yv;
                    sxx = fmaf(xv, xv, sxx);
                    syy = fmaf(yv, yv, syy);
                    sxy = fmaf(xv, yv, sxy);
                }
            }
            bx[t] = sx; by[t] = sy; bxx[t] = sxx; byy[t] = syy; bxy[t] = sxy;
        }
        v2f Bx  = {bx[0],  bx[1]};
        v2f By  = {by[0],  by[1]};
        v2f Bxx = {bxx[0], bxx[1]};
        v2f Byy = {byy[0], byy[1]};
        v2f Bxy = {bxy[0], bxy[1]};

        Sx  = __builtin_amdgcn_wmma_f32_16x16x4_f32(false, A, false, Bx,  (short)0, Sx,  false, false);
        Sy  = __builtin_amdgcn_wmma_f32_16x16x4_f32(false, A, false, By,  (short)0, Sy,  false, false);
        Sxx = __builtin_amdgcn_wmma_f32_16x16x4_f32(false, A, false, Bxx, (short)0, Sxx, false, false);
        Syy = __builtin_amdgcn_wmma_f32_16x16x4_f32(false, A, false, Byy, (short)0, Syy, false, false);
        Sxy = __builtin_amdgcn_wmma_f32_16x16x4_f32(false, A, false, Bxy, (short)0, Sxy, false, false);
    }

    // ---- SSIM formula on the 8 accumulator elements this lane owns ----
    const float inv_np   = 1.0f / 49.0f;
    const float cov_norm = 49.0f / 48.0f;
    const float C1 = 0.01f * 0.01f;
    const float C2 = 0.03f * 0.03f;

    float lsum = 0.0f;
    #pragma unroll
    for (int v = 0; v < 8; ++v) {
        const int orow = r0 + wr0 + v + 8 * half;   // C/D layout: M = v + 8*half
        const int ocol = c0 + wc0 + lcol;           // N = lane % 16
        if (orow < OUTD && ocol < OUTD) {
            float ux  = Sx[v]  * inv_np;
            float uy  = Sy[v]  * inv_np;
            float uxx = Sxx[v] * inv_np;
            float uyy = Syy[v] * inv_np;
            float uxy = Sxy[v] * inv_np;
            float vx  = cov_norm * (uxx - ux * ux);
            float vy  = cov_norm * (uyy - uy * uy);
            float vxy = cov_norm * (uxy - ux * uy);
            float A1  = 2.0f * ux * uy + C1;
            float A2  = 2.0f * vxy + C2;
            float B1  = ux * ux + uy * uy + C1;
            float B2  = vx + vy + C2;
            lsum += (A1 * A2) / (B1 * B2);
        }
    }

    // ---- wave32 reduction, then block reduction to one partial ----
    #pragma unroll
    for (int off = 16; off > 0; off >>= 1)
        lsum += __shfl_xor(lsum, off, 32);
    if (lane == 0) wsum[wave] = lsum;
    __syncthreads();
    if (tid == 0) {
        float s = 0.f;
        #pragma unroll
        for (int w = 0; w < 8; ++w) s += wsum[w];
        partial[bid] = s;
    }
}

__launch_bounds__(256)
__global__ void ssim_reduce_kernel(const float* __restrict__ partial,
                                   float* __restrict__ out) {
    __shared__ float buf[256];
    float s = 0.f;
    for (int i = threadIdx.x; i < NBLOCKS; i += 256) s += partial[i];
    buf[threadIdx.x] = s;
    __syncthreads();
    for (int stride = 128; stride > 0; stride >>= 1) {
        if (threadIdx.x < stride) buf[threadIdx.x] += buf[threadIdx.x + stride];
        __syncthreads();
    }
    if (threadIdx.x == 0) {
        float mean = buf[0] / ((float)NBATCH * (float)OUTD * (float)OUTD);
        out[0] = 1.0f - mean;
    }
}

extern "C" void kernel_launch(void* const* d_in, const int* in_sizes, int n_in,
                              void* d_out, int out_size, void* d_ws, size_t ws_size,
                              hipStream_t stream) {
    const float* img1 = (const float*)d_in[0];
    const float* img2 = (const float*)d_in[1];
    float* out = (float*)d_out;
    float* partial = (float*)d_ws;   // NBLOCKS floats = 32 KB scratch

    ssim_tile_kernel<<<NBLOCKS, 256, 0, stream>>>(img1, img2, partial);
    ssim_reduce_kernel<<<1, 256, 0, stream>>>(partial, out);
}